// FFTConvLean_36344013258829
// MI455X (gfx1250) — compile-verified
//
#include <hip/hip_runtime.h>

// ============================================================================
// S4-style FFT long convolution on MI455X (gfx1250):
//   y = irfft(rfft(x, 2L) * rfft(k, 2L))[..., :L] + x * D,  B=16 H=1024 L=4096
//
// - packed-real trick: one 4096-pt complex FFT per length-8192 real transform
// - 4096 = 16^3 radix-16 FFT; every radix-16 group is a 16x16 complex matmul
//   against the DFT-16 matrix, executed with V_WMMA_F32_16X16X4_F32
// - rfft(k) precomputed once into d_ws (33.6 MB -> lives in the 192 MB L2)
// - one workgroup (16 wave32) per (b,h) row, 34 KB LDS, fully in-place
// ============================================================================

typedef float v2f __attribute__((ext_vector_type(2)));
typedef float v8f __attribute__((ext_vector_type(8)));

namespace {

constexpr int   kL        = 4096;   // sequence length (complex FFT size too)
constexpr int   kH        = 1024;
constexpr int   kKfStride = 4104;   // 4097 complex bins per channel, padded
constexpr float kPi       = 3.14159265358979323846f;

__device__ __forceinline__ v8f wmma_f32(v2f a, v2f b, v8f c) {
  // D(16x16,f32) += A(16x4,f32) x B(4x16,f32)
  return __builtin_amdgcn_wmma_f32_16x16x4_f32(false, a, false, b, (short)0, c,
                                               false, false);
}

// one K=4 chunk of a complex 16x16x16 product:
//   cre += Ar*Br ; t += Ai*Bi (subtracted later) ; cim += Ar*Bi + Ai*Br
__device__ __forceinline__ void cmm_chunk(v8f& cre, v8f& cim, v8f& t,
                                          v2f ar, v2f ai, v2f br, v2f bi) {
  cre = wmma_f32(ar, br, cre);
  t   = wmma_f32(ai, bi, t);
  cim = wmma_f32(ar, bi, cim);
  cim = wmma_f32(ai, br, cim);
}

// ----------------------------------------------------------------------------
// In-place 4096-point complex DIT FFT, 16 waves cooperating, all butterfly
// groups executed as WMMA tile products against the DFT-16 matrix F16.
//   n = n1*256 + n2 ; k = k1 + 16*k2 ; k2 = k2a + 16*k2b
// ----------------------------------------------------------------------------
__device__ __forceinline__ void fft4096_wmma(float* zre, float* zim,
                                             const float* f16re,
                                             const float* f16im,
                                             int wave, int lane) {
  const int half = lane >> 4;   // 0..1   (K-halves of A / B operands)
  const int lr   = lane & 15;   // 0..15  (A row index == B/C column index)

  // -------- stage 1 : C = F16 * Zmat(16x256), + W4096^(n2*k1), in place ----
  {
    v8f cre = {}, cim = {}, t = {};
    const int cb = wave << 4;   // this wave's private 16-column block (n2)
#pragma unroll
    for (int kk = 0; kk < 4; ++kk) {
      const int r0 = 4 * kk + 2 * half;
      v2f ar, ai, br, bi;
      ar.x = f16re[lr * 16 + r0];     ar.y = f16re[lr * 16 + r0 + 1];
      ai.x = f16im[lr * 16 + r0];     ai.y = f16im[lr * 16 + r0 + 1];
      br.x = zre[r0 * 256 + cb + lr]; br.y = zre[(r0 + 1) * 256 + cb + lr];
      bi.x = zim[r0 * 256 + cb + lr]; bi.y = zim[(r0 + 1) * 256 + cb + lr];
      cmm_chunk(cre, cim, t, ar, ai, br, bi);
    }
#pragma unroll
    for (int v = 0; v < 8; ++v) cre[v] -= t[v];
#pragma unroll
    for (int v = 0; v < 8; ++v) {     // twiddle + in-place store
      const int k1 = v + 8 * half;
      const int n2 = cb + lr;
      float sn, cs;
      __sincosf(-2.0f * kPi * (float)(k1 * n2) * (1.0f / 4096.0f), &sn, &cs);
      zre[k1 * 256 + n2] = cre[v] * cs - cim[v] * sn;
      zim[k1 * 256 + n2] = cre[v] * sn + cim[v] * cs;
    }
  }
  __syncthreads();

  // -------- stage 2 : DFT-256 on row k1 = wave (two 16x16 tile products) ---
  const int rb = wave << 8;     // this wave's private row base
  v8f gre = {}, gim = {};
  {
    // E = F16 * Cm, Cm[p][q] = row[p*16+q]
    v8f ere = {}, eim = {}, t = {};
#pragma unroll
    for (int kk = 0; kk < 4; ++kk) {
      const int r0 = 4 * kk + 2 * half;
      v2f ar, ai, br, bi;
      ar.x = f16re[lr * 16 + r0];    ar.y = f16re[lr * 16 + r0 + 1];
      ai.x = f16im[lr * 16 + r0];    ai.y = f16im[lr * 16 + r0 + 1];
      br.x = zre[rb + r0 * 16 + lr]; br.y = zre[rb + (r0 + 1) * 16 + lr];
      bi.x = zim[rb + r0 * 16 + lr]; bi.y = zim[rb + (r0 + 1) * 16 + lr];
      cmm_chunk(ere, eim, t, ar, ai, br, bi);
    }
#pragma unroll
    for (int v = 0; v < 8; ++v) ere[v] -= t[v];
    // E' = E .* W256^(q*k2a), parked back into this wave's private row
    // (wave-synchronous: per-wave DS ops are in-order)
#pragma unroll
    for (int v = 0; v < 8; ++v) {
      const int k2a = v + 8 * half;
      float sn, cs;
      __sincosf(-2.0f * kPi * (float)(k2a * lr) * (1.0f / 256.0f), &sn, &cs);
      zre[rb + k2a * 16 + lr] = ere[v] * cs - eim[v] * sn;
      zim[rb + k2a * 16 + lr] = ere[v] * sn + eim[v] * cs;
    }
    // G^T = F16 * E'^T  (rows k2b, cols k2a)
    v8f t2 = {};
#pragma unroll
    for (int kk = 0; kk < 4; ++kk) {
      const int r0 = 4 * kk + 2 * half;   // r0 = q index
      v2f ar, ai, br, bi;
      ar.x = f16re[lr * 16 + r0];    ar.y = f16re[lr * 16 + r0 + 1];
      ai.x = f16im[lr * 16 + r0];    ai.y = f16im[lr * 16 + r0 + 1];
      br.x = zre[rb + lr * 16 + r0]; br.y = zre[rb + lr * 16 + r0 + 1];
      bi.x = zim[rb + lr * 16 + r0]; bi.y = zim[rb + lr * 16 + r0 + 1];
      cmm_chunk(gre, gim, t2, ar, ai, br, bi);
    }
#pragma unroll
    for (int v = 0; v < 8; ++v) gre[v] -= t2[v];
  }
  __syncthreads();   // everyone done reading their rows -> safe to scatter
#pragma unroll
  for (int v = 0; v < 8; ++v) {
    // Z[k1 + 16*k2a + 256*k2b] ; k1 = wave, k2a = lr, k2b = v + 8*half
    const int idx = wave + 16 * lr + 256 * (v + 8 * half);
    zre[idx] = gre[v];
    zim[idx] = gim[v];
  }
  __syncthreads();
}

__device__ __forceinline__ void init_f16_tables(float* f16re, float* f16im,
                                                int tid) {
  if (tid < 256) {
    const int r = tid >> 4, c = tid & 15;
    float sn, cs;
    __sincosf(-2.0f * kPi * (float)(r * c) * (1.0f / 16.0f), &sn, &cs);
    f16re[tid] = cs;
    f16im[tid] = sn;
  }
}

}  // namespace

// ----------------------------------------------------------------------------
// Kernel 1: Kf[h][k] = rfft(k_h zero-padded to 8192)[k], k = 0..4096, -> d_ws
// ----------------------------------------------------------------------------
__global__ __launch_bounds__(512) void fftconv_kf_kernel(
    const float* __restrict__ kin, float2* __restrict__ kf) {
  __shared__ float zre[4096], zim[4096];
  __shared__ float f16re[256], f16im[256];
  const int tid  = threadIdx.x;
  const int wave = tid >> 5, lane = tid & 31;
  const int h    = blockIdx.x;

  init_f16_tables(f16re, f16im, tid);
  const float2* k2p = (const float2*)(kin + (size_t)h * kL);
  for (int n = tid; n < 2048; n += 512) {        // pack z[n]=k[2n]+i*k[2n+1]
    const float2 v = k2p[n];
    zre[n] = v.x; zim[n] = v.y;
  }
  for (int n = 2048 + tid; n < 4096; n += 512) { zre[n] = 0.f; zim[n] = 0.f; }
  __syncthreads();

  fft4096_wmma(zre, zim, f16re, f16im, wave, lane);

  // untangle packed-real spectrum: S[k] = Ze[k] + W_8192^k * Zo[k]
  float2* kfrow = kf + (size_t)h * kKfStride;
  for (int k = tid; k <= 2048; k += 512) {
    const int m  = 4096 - k;
    const int mi = m & 4095;
    const float zkr = zre[k],  zki = zim[k];
    const float zmr = zre[mi], zmi = zim[mi];
    const float Zer = 0.5f * (zkr + zmr), Zei =  0.5f * (zki - zmi);
    const float Zor = 0.5f * (zki + zmi), Zoi = -0.5f * (zkr - zmr);
    float sn, cs;
    __sincosf(-kPi * (float)k * (1.0f / 4096.0f), &sn, &cs);
    kfrow[k] = make_float2(Zer + (cs * Zor - sn * Zoi),
                           Zei + (cs * Zoi + sn * Zor));
    if (k != 2048) {  // partner bin m (k==0 produces the Nyquist bin m=4096)
      const float Wmr = -cs, Wmi = sn;   // W_8192^(4096-k)
      kfrow[m] = make_float2( Zer + (Wmr * Zor + Wmi * Zoi),
                             -Zei + (Wmi * Zor - Wmr * Zoi));
    }
  }
}

// ----------------------------------------------------------------------------
// Kernel 2: per (b,h) row: FFT -> untangle*Kf->repack (in place) -> IFFT -> y
// ----------------------------------------------------------------------------
__global__ __launch_bounds__(512) void fftconv_main_kernel(
    const float* __restrict__ x, const float2* __restrict__ kf,
    const float* __restrict__ Dp, float* __restrict__ out) {
  __shared__ float zre[4096], zim[4096];
  __shared__ float f16re[256], f16im[256];
  const int tid  = threadIdx.x;
  const int wave = tid >> 5, lane = tid & 31;
  const int bh   = blockIdx.x;
  const int h    = bh & (kH - 1);

  init_f16_tables(f16re, f16im, tid);
  const float2* x2    = (const float2*)(x + (size_t)bh * kL);
  const float2* kfrow = kf + (size_t)h * kKfStride;
  __builtin_prefetch(&kfrow[tid], 0, 0);   // warm L2->WGP path for Kf

  for (int n = tid; n < 2048; n += 512) {        // pack z[n]=x[2n]+i*x[2n+1]
    const float2 v = x2[n];
    zre[n] = v.x; zim[n] = v.y;
  }
  for (int n = 2048 + tid; n < 4096; n += 512) { zre[n] = 0.f; zim[n] = 0.f; }
  __syncthreads();

  fft4096_wmma(zre, zim, f16re, f16im, wave, lane);   // forward FFT

  // spectral: untangle -> Y = S*Kf -> repack Q (conjugated for inverse),
  // processed as independent (k, 4096-k) pairs, fully in place
  for (int k = tid; k <= 2048; k += 512) {
    const int m  = 4096 - k;
    const int mi = m & 4095;
    const float zkr = zre[k],  zki = zim[k];
    const float zmr = zre[mi], zmi = zim[mi];
    const float Zer = 0.5f * (zkr + zmr), Zei =  0.5f * (zki - zmi);
    const float Zor = 0.5f * (zki + zmi), Zoi = -0.5f * (zkr - zmr);
    float sn, cs;
    __sincosf(-kPi * (float)k * (1.0f / 4096.0f), &sn, &cs);
    const float Skr = Zer + (cs * Zor - sn * Zoi);
    const float Ski = Zei + (cs * Zoi + sn * Zor);
    const float Wmr = -cs, Wmi = sn;
    const float Smr =  Zer + (Wmr * Zor + Wmi * Zoi);
    const float Smi = -Zei + (Wmi * Zor - Wmr * Zoi);

    const float2 kfk = kfrow[k];
    const float2 kfm = kfrow[m];
    const float Ykr = Skr * kfk.x - Ski * kfk.y;
    const float Yki = Skr * kfk.y + Ski * kfk.x;
    const float Ymr = Smr * kfm.x - Smi * kfm.y;
    const float Ymi = Smr * kfm.y + Smi * kfm.x;

    // Q[k] = Ye + i * e^{+i*pi*k/4096} * Yo   (store conj(Q) for inverse)
    const float Yer = 0.5f * (Ykr + Ymr), Yei = 0.5f * (Yki - Ymi);
    const float Yor = 0.5f * (Ykr - Ymr), Yoi = 0.5f * (Yki + Ymi);
    zre[k] =   Yer - (cs * Yoi - sn * Yor);
    zim[k] = -(Yei + (cs * Yor + sn * Yoi));
    if (k >= 1 && k < 2048) {                 // partner bin Q[m]
      const float Yemr = 0.5f * (Ymr + Ykr), Yemi = 0.5f * (Ymi - Yki);
      const float Yomr = 0.5f * (Ymr - Ykr), Yomi = 0.5f * (Ymi + Yki);
      zre[m] =   Yemr + ( cs * Yomi + sn * Yomr);
      zim[m] = -(Yemi + (-cs * Yomr + sn * Yomi));
    }
  }
  __syncthreads();

  fft4096_wmma(zre, zim, f16re, f16im, wave, lane);   // inverse via conj trick

  // y[2n] = Re q[n], y[2n+1] = Im q[n], q = conj(R)/4096 ; add skip x*D
  const float Dh  = Dp[h];
  const float inv = 1.0f / 4096.0f;
  float2* out2 = (float2*)(out + (size_t)bh * kL);
  for (int n = tid; n < 2048; n += 512) {
    const float2 xv = x2[n];
    float2 o;
    o.x =  zre[n] * inv + xv.x * Dh;
    o.y = -zim[n] * inv + xv.y * Dh;
    out2[n] = o;
  }
}

// ----------------------------------------------------------------------------
extern "C" void kernel_launch(void* const* d_in, const int* in_sizes, int n_in,
                              void* d_out, int out_size, void* d_ws,
                              size_t ws_size, hipStream_t stream) {
  (void)n_in; (void)out_size; (void)ws_size;
  const float* x   = (const float*)d_in[0];   // (B,H,L) f32
  const float* kin = (const float*)d_in[1];   // (1,H,L) f32
  const float* Dp  = (const float*)d_in[2];   // (1,H,1) f32
  float*  out = (float*)d_out;
  float2* kfw = (float2*)d_ws;                // H * 4104 complex bins, 33.6 MB

  const int rows = in_sizes[0] / kL;          // B*H = 16384

  fftconv_kf_kernel<<<dim3(kH), dim3(512), 0, stream>>>(kin, kfw);
  fftconv_main_kernel<<<dim3(rows), dim3(512), 0, stream>>>(x, kfw, Dp, out);
}